// UnifiedGNN_17592186044976
// MI455X (gfx1250) — compile-verified
//
#include <hip/hip_runtime.h>
#include <stdint.h>

// ---------------------------------------------------------------------------
// RGCN (2-layer, basis decomposition) for MI455X / gfx1250.
//  - GEMMs: V_WMMA_F32_16X16X4_F32, weights pre-swizzled to B-fragment order
//  - Weight staging: double-buffered global_load_async_to_lds_b128 (GVS mode,
//    SGPR base + 32-bit lane offset) overlapped with WMMA compute
//  - Edge scatter: float atomics (the true roofline limiter of this workload)
// ---------------------------------------------------------------------------

typedef __attribute__((ext_vector_type(2))) float v2f;
typedef __attribute__((ext_vector_type(8))) float v8f;

#define DFEAT   128
#define NTILES  8       // 128 / 16 output column tiles per wave
#define KCHUNK  64      // K staged per LDS round
#define PPAIR   144     // padded pair-row stride (pairs); half-wave b64 reads
                        // land on disjoint 32-bank windows
#define CHUNKF  8192    // floats per swizzled 64x128 weight chunk
#define BUFFLT  ((KCHUNK / 2) * PPAIR * 2)   // floats per LDS buffer (9216)

struct GemmArgs {
    const float* A[6];     // per-chunk A source ([*,128] row-major)
    int          acol[6];  // per-chunk starting column inside A source
    const float* W[6];     // per-chunk SWIZZLED weight chunk (8192 floats)
    int          nChunks;
    const float* bias;     // [128]
    float*       C;        // [nRows,128]
    int          nRows;
    int          relu;
};

// ---------------------------------------------------------------------------
// Pre-swizzle weights [K,128] row-major -> B-fragment order:
//   dst[(k>>6)*8192 + ((k>>1)&31)*256 + n*2 + (k&1)] = src[k*128 + n]
// so each (fr,n) holds the contiguous pair (W[2fr][n], W[2fr+1][n]).
// ---------------------------------------------------------------------------
__global__ __launch_bounds__(256) void rgcn_swizzle_w(
        const float* __restrict__ src, float* __restrict__ dst, int K) {
    const int t = blockIdx.x * blockDim.x + threadIdx.x;
    if (t >= K * DFEAT) return;
    const int k = t >> 7;
    const int n = t & 127;
    dst[((k >> 6) << 13) + (((k >> 1) & 31) << 8) + (n << 1) + (k & 1)] = src[t];
}

// ---------------------------------------------------------------------------
__global__ __launch_bounds__(256) void rgcn_gemm_wmma(GemmArgs g) {
    // two 36864-byte weight stage buffers (double buffered)
    __shared__ __align__(16) float lw[2][BUFFLT];

    const int tid  = threadIdx.x;
    const int lane = tid & 31;
    const int wave = tid >> 5;
    const int hi   = lane >> 4;      // 0: lanes 0-15, 1: lanes 16-31
    const int lo   = lane & 15;
    const int m0   = blockIdx.x * 128 + wave * 16;   // wave's first row
    const int arow = m0 + lo;                        // row this lane feeds A
    const bool arow_ok = (arow < g.nRows);
    const int arow_c   = arow_ok ? arow : 0;

    const unsigned ldsBase   = (unsigned)(uintptr_t)&lw[0][0];
    const unsigned bufStride = BUFFLT * sizeof(float);           // 36864
    // this thread's base inside a stage buffer: frag row tid>>6, unit tid&63
    const unsigned myLds  = (tid >> 6) * (PPAIR * 8) + (tid & 63) * 16;
    const unsigned myVoff = tid * 16;   // 32-bit GVS lane offset, invariant

    // stage one 32 KB swizzled weight chunk into LDS buffer `buf` (async)
    auto stage = [&](const float* wc, int buf) {
        const unsigned base = ldsBase + (unsigned)buf * bufStride + myLds;
        #pragma unroll
        for (int i = 0; i < 8; ++i) {
            const unsigned ldsAddr = base + i * (4 * PPAIR * 8);  // +4 frag rows
            const unsigned voff    = myVoff + i * 4096;
            asm volatile("global_load_async_to_lds_b128 %0, %1, %2"
                         :: "v"(ldsAddr), "v"(voff), "s"(wc) : "memory");
        }
    };

    v8f acc[NTILES] = {};   // zero-initialized accumulators

    // prologue: stage chunk 0
    stage(g.W[0], 0);
    asm volatile("s_wait_asynccnt 0x0" ::: "memory");
    __syncthreads();

    for (int c = 0; c < g.nChunks; ++c) {
        // kick off next chunk into the alternate buffer (overlaps compute)
        if (c + 1 < g.nChunks) stage(g.W[c + 1], (c + 1) & 1);
        if (c + 2 < g.nChunks) __builtin_prefetch(g.W[c + 2] + tid * 32, 0, 3);

        const float* lcur = &lw[c & 1][0];

        // A: 16x4 f32 fragment -- lane m holds row m;
        // VGPR0/1 = K {0,1} (lanes 0-15) or K {2,3} (lanes 16-31).
        const float* abase = g.A[c] + (size_t)arow_c * DFEAT + g.acol[c] + (hi << 1);

        #pragma unroll
        for (int k = 0; k < KCHUNK; k += 4) {
            v2f a;
            if (arow_ok) {
                const float* p = abase + k;
                a.x = p[0]; a.y = p[1];
            } else {
                a.x = 0.0f; a.y = 0.0f;
            }
            const int fr = (k >> 1) + hi;   // fragment row = (k + 2*hi)/2
            #pragma unroll
            for (int t = 0; t < NTILES; ++t) {
                const int n = t * 16 + lo;
                const v2f b = *(const v2f*)&lcur[(fr * PPAIR + n) * 2];
                acc[t] = __builtin_amdgcn_wmma_f32_16x16x4_f32(
                    /*neg_a=*/false, a, /*neg_b=*/false, b,
                    /*c_mod=*/(short)0, acc[t],
                    /*reuse_a=*/false, /*reuse_b=*/false);
            }
        }

        // next buffer fully landed + all waves done reading current buffer
        asm volatile("s_wait_asynccnt 0x0" ::: "memory");
        __syncthreads();
    }

    // ---- epilogue: C/D layout -> VGPR j holds rows j (lo half), j+8 (hi) --
    #pragma unroll
    for (int t = 0; t < NTILES; ++t) {
        const int col = t * 16 + lo;
        const float bv = g.bias[col];
        #pragma unroll
        for (int j = 0; j < 8; ++j) {
            const int row = m0 + j + (hi << 3);
            if (row < g.nRows) {
                float v = acc[t][j] + bv;
                if (g.relu) v = fmaxf(v, 0.0f);
                g.C[(size_t)row * DFEAT + col] = v;
            }
        }
    }
}

// ---------------------------------------------------------------------------
// Edge scatter: one wave32 per edge; each lane handles 4 feature floats.
// ---------------------------------------------------------------------------
__global__ __launch_bounds__(256) void rgcn_scatter(
        const float* __restrict__ h,
        const int*   __restrict__ esrc,   // [R*E] flat
        const int*   __restrict__ edst,   // [R*E] flat
        float* __restrict__ agg,          // [R*N,128]
        float* __restrict__ deg,          // [R*N]
        int RE, int Eper, int Nn) {
    const int gidx = blockIdx.x * blockDim.x + threadIdx.x;
    const int e    = gidx >> 5;
    const int lane = gidx & 31;
    if (e >= RE) return;

    const int r   = e / Eper;
    const int src = esrc[e];
    const int dst = edst[e];
    const size_t seg = (size_t)r * Nn + dst;

    const float4 v = *(const float4*)(h + (size_t)src * DFEAT + lane * 4);
    float* a = agg + seg * DFEAT + lane * 4;
    atomicAdd(a + 0, v.x);
    atomicAdd(a + 1, v.y);
    atomicAdd(a + 2, v.z);
    atomicAdd(a + 3, v.w);
    if (lane == 0) atomicAdd(deg + seg, 1.0f);
}

// ---------------------------------------------------------------------------
// Combine: aggB_b[n,:] = sum_r coeff[r,b] * agg[r,n,:] / max(deg[r,n],1)
// ---------------------------------------------------------------------------
__global__ __launch_bounds__(256) void rgcn_combine(
        const float* __restrict__ agg,    // [R*N,128]
        const float* __restrict__ deg,    // [R*N]
        const float* __restrict__ coeff,  // [R,2]
        float* __restrict__ b0,           // [N,128]
        float* __restrict__ b1,           // [N,128]
        int Nn) {
    const int gidx = blockIdx.x * blockDim.x + threadIdx.x;
    const int node = gidx >> 5;
    const int q    = gidx & 31;
    if (node >= Nn) return;

    float4 s0 = {0.f, 0.f, 0.f, 0.f};
    float4 s1 = {0.f, 0.f, 0.f, 0.f};
    #pragma unroll
    for (int r = 0; r < 3; ++r) {
        const size_t seg = (size_t)r * Nn + node;
        const float inv = 1.0f / fmaxf(deg[seg], 1.0f);
        const float4 a = *(const float4*)(agg + seg * DFEAT + q * 4);
        const float c0 = coeff[r * 2 + 0] * inv;
        const float c1 = coeff[r * 2 + 1] * inv;
        s0.x += c0 * a.x; s0.y += c0 * a.y; s0.z += c0 * a.z; s0.w += c0 * a.w;
        s1.x += c1 * a.x; s1.y += c1 * a.y; s1.z += c1 * a.z; s1.w += c1 * a.w;
    }
    *(float4*)(b0 + (size_t)node * DFEAT + q * 4) = s0;
    *(float4*)(b1 + (size_t)node * DFEAT + q * 4) = s1;
}

// ---------------------------------------------------------------------------
extern "C" void kernel_launch(void* const* d_in, const int* in_sizes, int n_in,
                              void* d_out, int out_size, void* d_ws, size_t ws_size,
                              hipStream_t stream) {
    const int Nn = 100000, R = 3, Eper = 500000;
    const int RE = R * Eper;

    const float* x      = (const float*)d_in[0];
    const int*   esrc   = (const int*)  d_in[1];
    const int*   edst   = (const int*)  d_in[2];
    const float* proj_w = (const float*)d_in[3];
    const float* proj_b = (const float*)d_in[4];
    const float* basis1 = (const float*)d_in[5];
    const float* coeff1 = (const float*)d_in[6];
    const float* bias1  = (const float*)d_in[7];
    const float* loop1  = (const float*)d_in[8];
    const float* basis2 = (const float*)d_in[9];
    const float* coeff2 = (const float*)d_in[10];
    const float* bias2  = (const float*)d_in[11];
    const float* loop2  = (const float*)d_in[12];

    // ---- workspace layout -------------------------------------------------
    char* ws = (char*)d_ws;
    size_t off = 0;
    auto take = [&](size_t bytes) -> float* {
        float* p = (float*)(ws + off);
        off += (bytes + 255) & ~(size_t)255;
        return p;
    };
    const size_t featBytes = (size_t)Nn * DFEAT * sizeof(float);
    float* h0    = take(featBytes);                       // proj output
    float* h1    = take(featBytes);                       // layer1 output
    float* agg   = take((size_t)R * featBytes);           // [R*N,128]
    float* deg   = take((size_t)R * Nn * sizeof(float));  // [R*N]
    float* b0    = take(featBytes);                       // aggB basis-0
    float* b1    = take(featBytes);                       // aggB basis-1
    float* swzP  = take((size_t)2 * CHUNKF * sizeof(float));  // proj (K=128)
    float* swzB1 = take((size_t)4 * CHUNKF * sizeof(float));  // basis1 (K=256)
    float* swzL1 = take((size_t)2 * CHUNKF * sizeof(float));  // loop1 (K=128)
    float* swzB2 = take((size_t)4 * CHUNKF * sizeof(float));  // basis2 (K=256)
    float* swzL2 = take((size_t)2 * CHUNKF * sizeof(float));  // loop2 (K=128)

    // ---- pre-swizzle all weight matrices into B-fragment order ------------
    rgcn_swizzle_w<<<(128 * DFEAT + 255) / 256, 256, 0, stream>>>(proj_w, swzP, 128);
    rgcn_swizzle_w<<<(256 * DFEAT + 255) / 256, 256, 0, stream>>>(basis1, swzB1, 256);
    rgcn_swizzle_w<<<(128 * DFEAT + 255) / 256, 256, 0, stream>>>(loop1,  swzL1, 128);
    rgcn_swizzle_w<<<(256 * DFEAT + 255) / 256, 256, 0, stream>>>(basis2, swzB2, 256);
    rgcn_swizzle_w<<<(128 * DFEAT + 255) / 256, 256, 0, stream>>>(loop2,  swzL2, 128);

    const int gemmBlocks    = (Nn + 127) / 128;
    const int scatterBlocks = (RE * 32 + 255) / 256;
    const int combineBlocks = (Nn * 32 + 255) / 256;

    // ---- projection: h0 = x @ proj_w + proj_b -----------------------------
    {
        GemmArgs g{};
        g.A[0] = x;  g.acol[0] = 0;   g.W[0] = swzP;
        g.A[1] = x;  g.acol[1] = 64;  g.W[1] = swzP + CHUNKF;
        g.nChunks = 2; g.bias = proj_b; g.C = h0; g.nRows = Nn; g.relu = 0;
        rgcn_gemm_wmma<<<gemmBlocks, 256, 0, stream>>>(g);
    }

    // ---- two RelGraphConv layers ------------------------------------------
    for (int layer = 0; layer < 2; ++layer) {
        const float* hin   = (layer == 0) ? h0 : h1;
        float*       hout  = (layer == 0) ? h1 : (float*)d_out;
        const float* swzB  = (layer == 0) ? swzB1 : swzB2;
        const float* swzL  = (layer == 0) ? swzL1 : swzL2;
        const float* coeff = (layer == 0) ? coeff1 : coeff2;
        const float* bias  = (layer == 0) ? bias1  : bias2;

        hipMemsetAsync(agg, 0, (size_t)R * featBytes, stream);
        hipMemsetAsync(deg, 0, (size_t)R * Nn * sizeof(float), stream);

        rgcn_scatter<<<scatterBlocks, 256, 0, stream>>>(
            hin, esrc, edst, agg, deg, RE, Eper, Nn);

        rgcn_combine<<<combineBlocks, 256, 0, stream>>>(
            agg, deg, coeff, b0, b1, Nn);

        // hout = relu( b0@basis[0] + b1@basis[1] + hin@loop + bias )
        GemmArgs g{};
        g.A[0] = b0;  g.acol[0] = 0;   g.W[0] = swzB;
        g.A[1] = b0;  g.acol[1] = 64;  g.W[1] = swzB + 1 * CHUNKF;
        g.A[2] = b1;  g.acol[2] = 0;   g.W[2] = swzB + 2 * CHUNKF;
        g.A[3] = b1;  g.acol[3] = 64;  g.W[3] = swzB + 3 * CHUNKF;
        g.A[4] = hin; g.acol[4] = 0;   g.W[4] = swzL;
        g.A[5] = hin; g.acol[5] = 64;  g.W[5] = swzL + 1 * CHUNKF;
        g.nChunks = 6; g.bias = bias; g.C = hout; g.nRows = Nn; g.relu = 1;
        rgcn_gemm_wmma<<<gemmBlocks, 256, 0, stream>>>(g);
    }
}